// Model_11879879542629
// MI455X (gfx1250) — compile-verified
//
#include <hip/hip_runtime.h>
#include <stdint.h>

typedef unsigned int uint32;
typedef uint32 v4u __attribute__((ext_vector_type(4)));
typedef int    v4i __attribute__((ext_vector_type(4)));
typedef int    v8i __attribute__((ext_vector_type(8)));

#define TILE_BYTES 65536u   // 64 KB per workgroup DMA tile

// ---------------------------------------------------------------------------
// Kernel 1: bulk copy x -> out using the Tensor Data Mover.
// One wave per block; each block moves one 64 KB tile global->LDS->global.
// Descriptor is built so that the "tensor" IS the tile (no OOB interaction):
//   data_size = 8B (code 3), tensor_dim0 = tile_dim0 = 8192 elements,
//   tensor_dim1 = tile_dim1 = 1, strides = 8192.
// ---------------------------------------------------------------------------
__global__ __launch_bounds__(32) void tdm_copy_kernel(const float* __restrict__ src,
                                                      float* __restrict__ dst)
{
    __shared__ __align__(128) unsigned char lbuf[TILE_BYTES];

    const uint64_t tile_off = (uint64_t)blockIdx.x * (uint64_t)TILE_BYTES;
    const uint32   lds_off  = (uint32)(uintptr_t)(&lbuf[0]);   // low 32 bits of flat LDS addr = LDS byte offset

    const uint64_t ga_src = (uint64_t)(uintptr_t)src + tile_off;
    const uint64_t ga_dst = (uint64_t)(uintptr_t)dst + tile_off;

    const uint32 elems = TILE_BYTES / 8u;   // 8192 x 8-byte elements

    // ---- D# group 1 (256 bits): sizes/strides, data_size=8B, no padding,
    //      no multicast, no iterate, no atomic-barrier.
    v8i g1;
    g1[0] = (int)0x00030000u;                          // workgroup_mask=0, data_size=3 (8B)
    g1[1] = (int)((elems & 0xffffu) << 16);            // tensor_dim0[15:0] at bits 63:48
    g1[2] = (int)((elems >> 16) | (1u << 16));         // tensor_dim0[31:16] | tensor_dim1[15:0]=1
    g1[3] = (int)((elems & 0xffffu) << 16);            // tensor_dim1[31:16]=0 | tile_dim0
    g1[4] = (int)1u;                                   // tile_dim1=1, tile_dim2=0
    g1[5] = (int)elems;                                // tensor_dim0_stride[31:0]
    g1[6] = (int)((elems & 0xffffu) << 16);            // stride0[47:32]=0 | tensor_dim1_stride[15:0]
    g1[7] = 0;                                         // tensor_dim1_stride[47:16]=0

    // ---- D# groups 2/3 unused (<=2D tensor): zero-filled; extra 6th-form
    //      operand (clang-23 signature) also zero-filled.
    v4i gz4 = {0, 0, 0, 0};
    v8i gz8 = {0, 0, 0, 0, 0, 0, 0, 0};

    // ---- D# group 0 for the load (global src -> LDS)
    v4u g0_load;
    g0_load[0] = 1u;                                                // count=1, user mode, no gather
    g0_load[1] = lds_off;                                           // lds_addr
    g0_load[2] = (uint32)(ga_src & 0xffffffffu);                    // global_addr[31:0]
    g0_load[3] = (uint32)(((ga_src >> 32) & 0x01ffffffu) | 0x80000000u); // addr[56:32] | type=2

    // ---- D# group 0 for the store (LDS -> global dst)
    v4u g0_store;
    g0_store[0] = 1u;
    g0_store[1] = lds_off;
    g0_store[2] = (uint32)(ga_dst & 0xffffffffu);
    g0_store[3] = (uint32)(((ga_dst >> 32) & 0x01ffffffu) | 0x80000000u);

    __builtin_amdgcn_tensor_load_to_lds(g0_load, g1, gz4, gz4, gz8, 0);
    __builtin_amdgcn_s_wait_tensorcnt(0);               // LDS tile resident
    __builtin_amdgcn_tensor_store_from_lds(g0_store, g1, gz4, gz4, gz8, 0);
    __builtin_amdgcn_s_wait_tensorcnt(0);               // keep LDS alive until DMA-out done
}

// ---------------------------------------------------------------------------
// Kernel 2: gather from img + scatter-add into out.
// One thread per (point, channel) pair: 64M threads, 1 gather + 1 atomic each.
// Lanes 0-15 / 16-31 of a wave share a point, so the four index loads
// coalesce to two addresses per wave. img (64MB) and out (64MB) are both
// L2-resident (192MB L2), so the random traffic stays on-die. The index
// streams (64MB total) are touched once -> non-temporal loads so they do
// not evict the L2-resident img/out working set.
// H = W = 1024 -> pure shift/OR addressing.
// ---------------------------------------------------------------------------
__global__ __launch_bounds__(256) void scatter_add_kernel(
    const float* __restrict__ img,
    const int*   __restrict__ index_x,
    const int*   __restrict__ index_y,
    const int*   __restrict__ proj_x,
    const int*   __restrict__ proj_y,
    float*       __restrict__ out,
    int npairs)
{
    int gid = blockIdx.x * 256 + threadIdx.x;
    if (gid >= npairs) return;

    uint32 l = (uint32)gid >> 4;
    uint32 c = (uint32)gid & 15u;

    uint32 sx = (uint32)__builtin_nontemporal_load(&proj_x[l]);   // column in W
    uint32 sy = (uint32)__builtin_nontemporal_load(&proj_y[l]);   // row in H
    uint32 dx = (uint32)__builtin_nontemporal_load(&index_x[l]);  // row in H
    uint32 dy = (uint32)__builtin_nontemporal_load(&index_y[l]);  // column in W

    float v = img[(c << 20) | (sy << 10) | sx];
    atomicAdd(&out[(c << 20) | (dx << 10) | dy], v);
}

// ---------------------------------------------------------------------------
extern "C" void kernel_launch(void* const* d_in, const int* in_sizes, int n_in,
                              void* d_out, int out_size, void* d_ws, size_t ws_size,
                              hipStream_t stream)
{
    const float* x       = (const float*)d_in[0];   // (1,16,1024,1024) f32
    const float* img     = (const float*)d_in[1];   // (16,1024,1024)   f32
    const int*   index_x = (const int*)  d_in[2];   // (L,1)
    const int*   index_y = (const int*)  d_in[3];   // (L,1)
    const int*   proj_x  = (const int*)  d_in[4];   // (L,)
    const int*   proj_y  = (const int*)  d_in[5];   // (L,)
    float*       out     = (float*)d_out;           // (16,1024,1024)   f32

    const int L = in_sizes[4];                      // 4,000,000 points

    // Phase 1: out = x   (64 MB, 1024 x 64KB TDM tiles; sizes divide exactly)
    const uint64_t total_bytes = (uint64_t)out_size * 4ull;
    const uint32_t ntiles = (uint32_t)(total_bytes / TILE_BYTES);
    tdm_copy_kernel<<<ntiles, 32, 0, stream>>>(x, out);

    // Phase 2: out[c, ix, iy] += img[c, py, px]  (stream-ordered after copy)
    const int npairs = L * 16;
    const int nblocks = (npairs + 255) / 256;
    scatter_add_kernel<<<nblocks, 256, 0, stream>>>(img, index_x, index_y,
                                                    proj_x, proj_y, out, npairs);
}